// ContextQueryAttention_22952305230013
// MI455X (gfx1250) — compile-verified
//
#include <hip/hip_runtime.h>

typedef float v2f __attribute__((ext_vector_type(2)));
typedef float v8f __attribute__((ext_vector_type(8)));

#define NEG_INF_F (-1000000000.0f)

constexpr int Bn = 64, Cn = 1024, Qn = 128, Dn = 128;
constexpr int P = 132;            // padded LDS pitch (dwords): 4n-stepped banks, conflict-free WMMA fragment loads
constexpr int CHUNK = 128;        // context rows per chunk
constexpr int NCHUNK = Cn / CHUNK;
constexpr int OUTW = 4 * Dn;      // 512
// LDS layout (floats)
constexpr int SM_FLOATS = 3 * (128 * P) + 384 + 128 + 128 + 1024 + 256 + 128;

__global__ __launch_bounds__(256, 1)
void cqa_kernel(const float* __restrict__ ctx_g,
                const float* __restrict__ qry_g,
                const float* __restrict__ W_g,
                const float* __restrict__ qmask_g,
                float* __restrict__ out_g)
{
    extern __shared__ float smem[];
    float* q_lds   = smem;                 // Qn * P
    float* ctx_lds = q_lds   + Qn * P;     // CHUNK * P
    float* sc_lds  = ctx_lds + CHUNK * P;  // CHUNK * P (scores, then c2q staging)
    float* w_lds   = sc_lds  + CHUNK * P;  // 384 : w_c | w_q | w_s
    float* qbias   = w_lds   + 384;        // Qn  : q_term + mask bias
    float* cterm   = qbias   + Qn;         // CHUNK
    float* m_all   = cterm   + CHUNK;      // Cn  : row maxima, then exp(p)
    float* red     = m_all   + Cn;         // 256 : reduction scratch
    float* q2c     = red     + 256;        // Dn

    const int b    = blockIdx.x;
    const int tid  = threadIdx.x;
    const int wv   = tid >> 5;     // wave 0..7
    const int lane = tid & 31;
    const int lh   = lane >> 4;    // lane half (K/M split in WMMA layouts)
    const int ln   = lane & 15;

    const float* ctxB = ctx_g + (size_t)b * Cn * Dn;
    const float* qryB = qry_g + (size_t)b * Qn * Dn;
    float*       outB = out_g + (size_t)b * Cn * OUTW;

    // ---- stage W and query; compute qbias[q] = q_term + (1-mask)*NEG_INF ----
    for (int i = tid; i < 3 * Dn; i += 256) w_lds[i] = W_g[i];
    for (int r = wv; r < Qn; r += 8) {
        const int c4 = lane * 4;
        const float4 v = *reinterpret_cast<const float4*>(qryB + (size_t)r * Dn + c4);
        *reinterpret_cast<float4*>(q_lds + r * P + c4) = v;
    }
    __syncthreads();
    if (tid < Qn) {
        float s = 0.f;
        for (int d = 0; d < Dn; ++d) s += q_lds[tid * P + d] * w_lds[Dn + d];
        qbias[tid] = s + (1.0f - qmask_g[b * Qn + tid]) * NEG_INF_F;
    }
    __syncthreads();

    const int rm = wv * 16;  // this wave's row base within the chunk

    for (int ch = 0; ch < NCHUNK; ++ch) {
        const int cc = ch * CHUNK;

        // ---- load context chunk ----
        for (int r = wv; r < CHUNK; r += 8) {
            const int c4 = lane * 4;
            const float4 v = *reinterpret_cast<const float4*>(ctxB + (size_t)(cc + r) * Dn + c4);
            *reinterpret_cast<float4*>(ctx_lds + r * P + c4) = v;
        }
        __syncthreads();
        if (tid < CHUNK) {
            float s = 0.f;
            for (int d = 0; d < Dn; ++d) s += ctx_lds[tid * P + d] * w_lds[d];
            cterm[tid] = s;
        }
        __syncthreads();

        // ---- S = (ctx * w_s) @ query^T : M=chunk rows, N=q, K=d ----
        v8f acc[8];
        #pragma unroll
        for (int t = 0; t < 8; ++t) acc[t] = v8f{0.f,0.f,0.f,0.f,0.f,0.f,0.f,0.f};

        for (int kb = 0; kb < Dn; kb += 4) {
            const int k0 = kb + 2 * lh;
            v2f a = *reinterpret_cast<const v2f*>(ctx_lds + (rm + ln) * P + k0);
            a.x *= w_lds[2 * Dn + k0];
            a.y *= w_lds[2 * Dn + k0 + 1];
            #pragma unroll
            for (int t = 0; t < 8; ++t) {
                // B[k][n] = query[n][k]
                const v2f bf = *reinterpret_cast<const v2f*>(q_lds + (t * 16 + ln) * P + k0);
                acc[t] = __builtin_amdgcn_wmma_f32_16x16x4_f32(
                    false, a, false, bf, (short)0, acc[t], false, false);
            }
        }

        // ---- bias + row softmax (C-layout: VGPR r holds rows r / r+8 per half) ----
        #pragma unroll
        for (int r = 0; r < 8; ++r) {
            const int row = rm + r + 8 * lh;
            const float cb = cterm[row];
            float sv[8];
            float mx = -3.4e38f;
            #pragma unroll
            for (int t = 0; t < 8; ++t) {
                const float s = acc[t][r] + cb + qbias[t * 16 + ln];
                sv[t] = s;
                mx = fmaxf(mx, s);
            }
            #pragma unroll
            for (int off = 1; off < 16; off <<= 1)
                mx = fmaxf(mx, __shfl_xor(mx, off, 32));
            if (ln == 0) m_all[cc + row] = mx;
            float sum = 0.f;
            #pragma unroll
            for (int t = 0; t < 8; ++t) { sv[t] = __expf(sv[t] - mx); sum += sv[t]; }
            #pragma unroll
            for (int off = 1; off < 16; off <<= 1)
                sum += __shfl_xor(sum, off, 32);
            const float inv = 1.0f / sum;
            #pragma unroll
            for (int t = 0; t < 8; ++t)
                sc_lds[row * P + t * 16 + ln] = sv[t] * inv;
        }
        __syncthreads();

        // ---- c2q = scores @ query : M=chunk rows, N=d, K=q ----
        v8f acc2[8];
        #pragma unroll
        for (int t = 0; t < 8; ++t) acc2[t] = v8f{0.f,0.f,0.f,0.f,0.f,0.f,0.f,0.f};

        for (int kb = 0; kb < Qn; kb += 4) {
            const int k0 = kb + 2 * lh;
            const v2f a = *reinterpret_cast<const v2f*>(sc_lds + (rm + ln) * P + k0);
            #pragma unroll
            for (int t = 0; t < 8; ++t) {
                v2f bf;
                bf.x = q_lds[k0 * P + t * 16 + ln];        // B[k][n] = query[k][n]
                bf.y = q_lds[(k0 + 1) * P + t * 16 + ln];
                acc2[t] = __builtin_amdgcn_wmma_f32_16x16x4_f32(
                    false, a, false, bf, (short)0, acc2[t], false, false);
            }
        }
        __syncthreads();  // everyone done reading scores before we overwrite

        #pragma unroll
        for (int r = 0; r < 8; ++r) {
            const int row = rm + r + 8 * lh;
            #pragma unroll
            for (int t = 0; t < 8; ++t)
                sc_lds[row * P + t * 16 + ln] = acc2[t][r];
        }
        __syncthreads();

        // ---- write out blocks 0..2: [ctx | c2q | ctx*c2q] ----
        for (int r = wv; r < CHUNK; r += 8) {
            const int c4 = lane * 4;
            const float4 cv = *reinterpret_cast<const float4*>(ctx_lds + r * P + c4);
            const float4 qv = *reinterpret_cast<const float4*>(sc_lds + r * P + c4);
            float* op = outB + (size_t)(cc + r) * OUTW;
            *reinterpret_cast<float4*>(op + c4) = cv;
            *reinterpret_cast<float4*>(op + Dn + c4) = qv;
            const float4 pv = make_float4(cv.x * qv.x, cv.y * qv.y, cv.z * qv.z, cv.w * qv.w);
            *reinterpret_cast<float4*>(op + 2 * Dn + c4) = pv;
        }
        __syncthreads();  // ctx_lds / sc_lds reused next chunk
    }

    // ---- q2c: softmax over c of m_all, then matvec against (L2-hot) context ----
    {
        float mx = -3.4e38f;
        for (int i = tid; i < Cn; i += 256) mx = fmaxf(mx, m_all[i]);
        red[tid] = mx;
        __syncthreads();
        for (int s = 128; s > 0; s >>= 1) {
            if (tid < s) red[tid] = fmaxf(red[tid], red[tid + s]);
            __syncthreads();
        }
        const float gmax = red[0];
        __syncthreads();
        float z = 0.f;
        for (int i = tid; i < Cn; i += 256) {
            const float p = __expf(m_all[i] - gmax);
            m_all[i] = p;
            z += p;
        }
        red[tid] = z;
        __syncthreads();
        for (int s = 128; s > 0; s >>= 1) {
            if (tid < s) red[tid] += red[tid + s];
            __syncthreads();
        }
        const float invZ = 1.0f / red[0];
        __syncthreads();

        const int d    = tid & 127;
        const int half = tid >> 7;
        float acc = 0.f;
        for (int c = half * 512; c < half * 512 + 512; ++c)
            acc += m_all[c] * ctxB[(size_t)c * Dn + d];
        red[tid] = acc;
        __syncthreads();
        if (tid < Dn) q2c[tid] = (red[tid] + red[tid + 128]) * invZ;
        __syncthreads();
    }

    // ---- block 3: ctx * q2c (context re-read hits L2) ----
    for (int idx = tid; idx < Cn * 32; idx += 256) {
        const int r  = idx >> 5;
        const int c4 = (idx & 31) * 4;
        const float4 cv = *reinterpret_cast<const float4*>(ctxB + (size_t)r * Dn + c4);
        const float4 qv = *reinterpret_cast<const float4*>(q2c + c4);
        const float4 pv = make_float4(cv.x * qv.x, cv.y * qv.y, cv.z * qv.z, cv.w * qv.w);
        *reinterpret_cast<float4*>(outB + (size_t)r * OUTW + 3 * Dn + c4) = pv;
    }
}

extern "C" void kernel_launch(void* const* d_in, const int* in_sizes, int n_in,
                              void* d_out, int out_size, void* d_ws, size_t ws_size,
                              hipStream_t stream) {
    const float* ctx   = (const float*)d_in[0];  // (64,1024,128) f32
    const float* qry   = (const float*)d_in[1];  // (64,128,128)  f32
    const float* W     = (const float*)d_in[2];  // (384,1)       f32
    const float* qmask = (const float*)d_in[3];  // (64,128)      f32
    float* out = (float*)d_out;                  // (64,1024,512) f32

    const size_t smem_bytes = (size_t)SM_FLOATS * sizeof(float);  // ~206 KB, < 320 KB/WGP
    hipFuncSetAttribute(reinterpret_cast<const void*>(cqa_kernel),
                        hipFuncAttributeMaxDynamicSharedMemorySize, (int)smem_bytes);
    cqa_kernel<<<dim3(Bn), dim3(256), smem_bytes, stream>>>(ctx, qry, W, qmask, out);
}